// SinglePositionalRelativeEmbedding_45904610459828
// MI455X (gfx1250) — compile-verified
//
#include <hip/hip_runtime.h>

// Problem geometry (fixed by the reference)
#define HDIM  64                    // h = w = sqrt(4096)
#define CDIM  320
#define BATCH 16
#define NCOL  (HDIM * CDIM)         // 20480 : pass-1 N dimension (j,c flattened)
#define LC    (HDIM * HDIM * CDIM)  // 1310720 floats per batch slice

typedef __attribute__((ext_vector_type(2))) float v2f;
typedef __attribute__((ext_vector_type(8))) float v8f;

// Band matrix A[row][k] = 1 iff max(0,row-3) <= k < min(64,row+3).
// k is always in [0,64) at call sites, so the clip is implicit.
__device__ __forceinline__ float band(int row, int k) {
    return (k >= row - 3 && k < row + 3) ? 1.0f : 0.0f;
}
// r(i) = number of valid window rows at position i (rowsum of A)
__device__ __forceinline__ float rcount(int i) {
    int top = i - 3; if (top < 0) top = 0;
    int bot = i + 3; if (bot > HDIM) bot = HDIM;
    return (float)(bot - top);
}

// ---------------------------------------------------------------------------
// Pass 1:  T(64 x 20480) = A(64x64) @ PE(64x20480)     [row box-sum]
// One wave computes one 16x16 tile; K stepped by 4 (16 WMMAs per tile).
// A-fragment is synthesized from the band predicate (zero loads for A).
// f32 WMMA layouts (ISA 7.12.2):
//   A 16x4 : lanes 0-15 -> M=lane, VGPR{0,1}=K{0,1}; lanes 16-31 -> K{2,3}
//   B 4x16 : lanes 0-15 -> N=lane, VGPR{0,1}=K{0,1}; lanes 16-31 -> K{2,3}
//   D 16x16: VGPR v, lanes 0-15 -> M=v, N=lane; lanes 16-31 -> M=v+8
// ---------------------------------------------------------------------------
__global__ void __launch_bounds__(128)
rowsum_wmma_kernel(const float* __restrict__ pe, float* __restrict__ t) {
    const int lane = threadIdx.x & 31;
    const int hh   = lane >> 4;       // lane half (selects K pair)
    const int lm   = lane & 15;
    const int tile = blockIdx.x * (blockDim.x >> 5) + (threadIdx.x >> 5);
    const int mtile = tile / (NCOL / 16);   // 0..3
    const int ntile = tile % (NCOL / 16);   // 0..1279
    const int row = mtile * 16 + lm;        // A row for this lane
    const int col = ntile * 16 + lm;        // B/D column for this lane

    v8f acc = {};
#pragma unroll
    for (int kk = 0; kk < 16; ++kk) {
        const int k0 = kk * 4 + hh * 2;
        v2f a; a.x = band(row, k0); a.y = band(row, k0 + 1);
        const float* bp = pe + (size_t)k0 * NCOL + col;
        v2f b; b.x = bp[0]; b.y = bp[NCOL];
        acc = __builtin_amdgcn_wmma_f32_16x16x4_f32(
            /*neg_a=*/false, a, /*neg_b=*/false, b,
            /*c_mod=*/(short)0, acc, /*reuse_a=*/false, /*reuse_b=*/false);
    }
#pragma unroll
    for (int v = 0; v < 8; ++v) {
        const int m = mtile * 16 + v + hh * 8;
        t[(size_t)m * NCOL + col] = acc[v];
    }
}

// ---------------------------------------------------------------------------
// Pass 2: per row-slab i: box_i(64x320) = A(64x64) @ T_i(64x320)
// Fused epilogue: rel[i,j,c] = r(i)*r(j) * pe[i,j,c] - box[i,j,c]
// ---------------------------------------------------------------------------
__global__ void __launch_bounds__(128)
colsum_rel_wmma_kernel(const float* __restrict__ pe, const float* __restrict__ t,
                       float* __restrict__ rel) {
    const int lane = threadIdx.x & 31;
    const int hh   = lane >> 4;
    const int lm   = lane & 15;
    const int tile = blockIdx.x * (blockDim.x >> 5) + (threadIdx.x >> 5);
    const int islab = tile / 80;            // grid row i, 0..63
    const int rem   = tile % 80;
    const int mtile = rem / 20;             // j tile, 0..3
    const int ntile = rem % 20;             // c tile, 0..19
    const int jrow = mtile * 16 + lm;
    const int c    = ntile * 16 + lm;
    const float* tb = t + (size_t)islab * NCOL;  // T_i, rows jj stride CDIM

    v8f acc = {};
#pragma unroll
    for (int kk = 0; kk < 16; ++kk) {
        const int k0 = kk * 4 + hh * 2;
        v2f a; a.x = band(jrow, k0); a.y = band(jrow, k0 + 1);
        const float* bp = tb + (size_t)k0 * CDIM + c;
        v2f b; b.x = bp[0]; b.y = bp[CDIM];
        acc = __builtin_amdgcn_wmma_f32_16x16x4_f32(
            false, a, false, b, (short)0, acc, false, false);
    }

    const float ri = rcount(islab);
#pragma unroll
    for (int v = 0; v < 8; ++v) {
        const int j = mtile * 16 + v + hh * 8;
        const float cnt = ri * rcount(j);
        const size_t idx = ((size_t)islab * HDIM + j) * CDIM + c;
        rel[idx] = cnt * pe[idx] - acc[v];
    }
}

// ---------------------------------------------------------------------------
// Pass 3: out[b, :] = x[b, :] + rel[:]   (pure streaming, float4, ~173 MB)
// grid = (LC/4/256, BATCH); rel (5.2 MB) stays resident in the 192 MB L2.
// ---------------------------------------------------------------------------
__global__ void __launch_bounds__(256)
add_bcast_kernel(const float* __restrict__ x, const float* __restrict__ rel,
                 float* __restrict__ out) {
    const size_t i = (size_t)blockIdx.x * blockDim.x + threadIdx.x; // 0..LC/4-1
    const size_t b = blockIdx.y;
    const float4 rv = ((const float4*)rel)[i];
    const float4 xv = ((const float4*)x)[b * (size_t)(LC / 4) + i];
    float4 o;
    o.x = xv.x + rv.x; o.y = xv.y + rv.y;
    o.z = xv.z + rv.z; o.w = xv.w + rv.w;
    ((float4*)out)[b * (size_t)(LC / 4) + i] = o;
}

extern "C" void kernel_launch(void* const* d_in, const int* in_sizes, int n_in,
                              void* d_out, int out_size, void* d_ws, size_t ws_size,
                              hipStream_t stream) {
    const float* x  = (const float*)d_in[0];   // (16, 4096, 320) f32
    const float* pe = (const float*)d_in[1];   // (1, 4096, 320)  f32
    float* out = (float*)d_out;

    // Stage T (row box-sums, 5.24 MB) in the front of d_out — pass 3 fully
    // overwrites it. rel (5.24 MB) lives in the workspace.
    float* t   = out;
    float* rel = (float*)d_ws;

    // Pass 1: 4 x 1280 tiles, 4 waves/block -> 1280 blocks of 128 threads.
    rowsum_wmma_kernel<<<1280, 128, 0, stream>>>(pe, t);
    // Pass 2: 64 slabs x 80 tiles = 5120 tiles -> same launch shape.
    colsum_rel_wmma_kernel<<<1280, 128, 0, stream>>>(pe, t, rel);
    // Pass 3: broadcast add, one float4 per thread.
    dim3 g3(LC / 4 / 256, BATCH);
    add_bcast_kernel<<<g3, 256, 0, stream>>>(x, rel, out);
}